// Basicgate_cvf_66726611910701
// MI455X (gfx1250) — compile-verified
//
#include <hip/hip_runtime.h>
#include <hip/hip_bf16.h>

#define H_DIM 160
#define W_DIM 480
#define HW (H_DIM * W_DIM)            // 76800
#define GW (W_DIM + 1)                // 481
#define GH (H_DIM + 1)                // 161
#define GCELLS (GW * GH)              // 77441
#define NP 120000
#define C0 64
#define C1 128
#define C2 256
#define CIMG 256
#define NPIX 32                       // pixels per GEMM block (2 N-tiles)

typedef __attribute__((ext_vector_type(2))) float v2f;
typedef __attribute__((ext_vector_type(8))) float v8f;

// ---------------------------------------------------------------- init idx
__global__ __launch_bounds__(256) void init_idx_kernel(int* __restrict__ idx) {
    int t = blockIdx.x * blockDim.x + threadIdx.x;
    if (t < 3 * GCELLS) idx[t] = -1;
}

// ---------------------------------------------------------------- scatter (last-write-wins via atomicMax on point index)
__global__ __launch_bounds__(256) void scatter_kernel(const int* __restrict__ g0,
                                                      const int* __restrict__ g1,
                                                      const int* __restrict__ g2,
                                                      int* __restrict__ idx) {
    int t = blockIdx.x * blockDim.x + threadIdx.x;
    if (t >= 3 * NP) return;
    int level = t / NP;
    int i = t - level * NP;
    const int* g = (level == 0) ? g0 : ((level == 1) ? g1 : g2);
    int x = g[2 * i];      // col in [0, W]
    int y = g[2 * i + 1];  // row in [0, H]
    atomicMax(&idx[level * GCELLS + y * GW + x], i);
}

// ---------------------------------------------------------------- enhanced = img_feat * seg_prob[1]
__global__ __launch_bounds__(256) void enhanced_kernel(const float* __restrict__ img_feat,
                                                       const float* __restrict__ seg_prob,
                                                       float* __restrict__ enh) {
    int t = blockIdx.x * blockDim.x + threadIdx.x;  // float4 index
    size_t e = (size_t)t * 4;
    if (e >= (size_t)CIMG * HW) return;
    int p = (int)(e % HW);
    float4 a = *(const float4*)(img_feat + e);
    float4 s = *(const float4*)(seg_prob + HW + p);  // channel 1 of [2,H,W]
    float4 o;
    o.x = a.x * s.x; o.y = a.y * s.y; o.z = a.z * s.z; o.w = a.w * s.w;
    *(float4*)(enh + e) = o;
}

// ---------------------------------------------------------------- pt_img GEMM (WMMA f32 16x16x4)
// One workgroup = 32 pixels (2 N-tiles; W%32==0 so no row crossing).
// 8 waves; each wave computes 2 M-tiles x 2 N-tiles. A fragments (weights)
// are loaded once per (k, M-tile) and reused for both N-tiles, halving the
// L2 weight-fetch traffic relative to a 16-pixel block.
__global__ __launch_bounds__(256) void ptimg_wmma_kernel(const float* __restrict__ vf0,
                                                         const float* __restrict__ vf1,
                                                         const float* __restrict__ vf2,
                                                         const int* __restrict__ idx,
                                                         const float* __restrict__ w0,
                                                         const float* __restrict__ b0,
                                                         const float* __restrict__ w1,
                                                         const float* __restrict__ b1,
                                                         float* __restrict__ pt_img) {
    __shared__ float lv0[C0 * NPIX];    //  8 KB
    __shared__ float lv1[C1 * NPIX];    // 16 KB
    __shared__ float lv2[C2 * NPIX];    // 32 KB

    const int t = threadIdx.x;
    const int pbase = blockIdx.x * NPIX;        // linear pixel base
    const int h = pbase / W_DIM;
    const int wbase = pbase - h * W_DIM;

    // ---- gather phase: 8 threads per pixel, float4 chunks, [channel][pixel] LDS layout
    {
        const int pi  = t & (NPIX - 1);         // 0..31
        const int sub = t >> 5;                 // 0..7
        const int cell = h * GW + (wbase + pi);
        const int i0 = idx[cell];
        const int i1 = idx[GCELLS + cell];
        const int i2 = idx[2 * GCELLS + cell];

        const float4 z4 = make_float4(0.f, 0.f, 0.f, 0.f);
#pragma unroll
        for (int j = 0; j < 2; ++j) {           // v0: 16 float4 per pixel
            int c4 = sub * 2 + j;
            float4 v = (i0 >= 0) ? ((const float4*)(vf0 + (size_t)i0 * C0))[c4] : z4;
            lv0[(c4 * 4 + 0) * NPIX + pi] = v.x;
            lv0[(c4 * 4 + 1) * NPIX + pi] = v.y;
            lv0[(c4 * 4 + 2) * NPIX + pi] = v.z;
            lv0[(c4 * 4 + 3) * NPIX + pi] = v.w;
        }
#pragma unroll
        for (int j = 0; j < 4; ++j) {           // v1: 32 float4 per pixel
            int c4 = sub * 4 + j;
            float4 v = (i1 >= 0) ? ((const float4*)(vf1 + (size_t)i1 * C1))[c4] : z4;
            lv1[(c4 * 4 + 0) * NPIX + pi] = v.x;
            lv1[(c4 * 4 + 1) * NPIX + pi] = v.y;
            lv1[(c4 * 4 + 2) * NPIX + pi] = v.z;
            lv1[(c4 * 4 + 3) * NPIX + pi] = v.w;
        }
#pragma unroll
        for (int j = 0; j < 8; ++j) {           // v2: 64 float4 per pixel
            int c4 = sub * 8 + j;
            float4 v = (i2 >= 0) ? ((const float4*)(vf2 + (size_t)i2 * C2))[c4] : z4;
            lv2[(c4 * 4 + 0) * NPIX + pi] = v.x;
            lv2[(c4 * 4 + 1) * NPIX + pi] = v.y;
            lv2[(c4 * 4 + 2) * NPIX + pi] = v.z;
            lv2[(c4 * 4 + 3) * NPIX + pi] = v.w;
        }
    }
    __syncthreads();

    // ---- WMMA phase
    const int lane = t & 31;
    const int wave = t >> 5;                    // 0..7 -> 2 M-tiles each
    const int hi = lane >> 4;                   // 0/1: K half for A/B fragments, M half for C/D
    const int n  = lane & 15;                   // N index (B/C/D); also M-within-tile for A

    v8f acc[2][2];                              // [mt][nt]
#pragma unroll
    for (int mt = 0; mt < 2; ++mt) {
        const int mbase = (wave * 2 + mt) * 16;
#pragma unroll
        for (int r = 0; r < 8; ++r) {
            int m = mbase + r + 8 * hi;
            float bias = b0[m] + b1[m];         // dense bias (added at every pixel)
            acc[mt][0][r] = bias;
            acc[mt][1][r] = bias;
        }
    }

    // level 0: K = 64
#pragma unroll 4
    for (int k = 0; k < C0; k += 4) {
        v2f bf[2];
#pragma unroll
        for (int nt = 0; nt < 2; ++nt) {
            bf[nt].x = lv0[(k + 0 + 2 * hi) * NPIX + nt * 16 + n];  // B VGPR0: K rows {k, k+2}
            bf[nt].y = lv0[(k + 1 + 2 * hi) * NPIX + nt * 16 + n];  // B VGPR1: K rows {k+1, k+3}
        }
#pragma unroll
        for (int mt = 0; mt < 2; ++mt) {
            const int m = (wave * 2 + mt) * 16 + n;                 // A: M = lane&15
            const float* wp = w0 + (size_t)m * C0 + k + 2 * hi;
            v2f a; a.x = wp[0]; a.y = wp[1];                        // A K rows {k+2hi, k+2hi+1}
            acc[mt][0] = __builtin_amdgcn_wmma_f32_16x16x4_f32(
                false, a, false, bf[0], (short)0, acc[mt][0], false, false);
            acc[mt][1] = __builtin_amdgcn_wmma_f32_16x16x4_f32(
                false, a, false, bf[1], (short)0, acc[mt][1], false, false);
        }
    }

    // level 1: K = 128
#pragma unroll 4
    for (int k = 0; k < C1; k += 4) {
        v2f bf[2];
#pragma unroll
        for (int nt = 0; nt < 2; ++nt) {
            bf[nt].x = lv1[(k + 0 + 2 * hi) * NPIX + nt * 16 + n];
            bf[nt].y = lv1[(k + 1 + 2 * hi) * NPIX + nt * 16 + n];
        }
#pragma unroll
        for (int mt = 0; mt < 2; ++mt) {
            const int m = (wave * 2 + mt) * 16 + n;
            const float* wp = w1 + (size_t)m * C1 + k + 2 * hi;
            v2f a; a.x = wp[0]; a.y = wp[1];
            acc[mt][0] = __builtin_amdgcn_wmma_f32_16x16x4_f32(
                false, a, false, bf[0], (short)0, acc[mt][0], false, false);
            acc[mt][1] = __builtin_amdgcn_wmma_f32_16x16x4_f32(
                false, a, false, bf[1], (short)0, acc[mt][1], false, false);
        }
    }

    // level 2 (identity) + store
#pragma unroll
    for (int mt = 0; mt < 2; ++mt) {
        const int mbase = (wave * 2 + mt) * 16;
#pragma unroll
        for (int nt = 0; nt < 2; ++nt) {
#pragma unroll
            for (int r = 0; r < 8; ++r) {
                int m = mbase + r + 8 * hi;
                float val = acc[mt][nt][r] + lv2[m * NPIX + nt * 16 + n];
                pt_img[(size_t)m * HW + pbase + nt * 16 + n] = val;
            }
        }
    }
}

// ---------------------------------------------------------------- 3x3 conv (512 -> 1) + sigmoid
#define TW 32
#define TH 8
__global__ __launch_bounds__(256) void conv_attn_kernel(const float* __restrict__ enh,
                                                        const float* __restrict__ pt_img,
                                                        const float* __restrict__ sbw,
                                                        const float* __restrict__ sbb,
                                                        float* __restrict__ attn) {
    __shared__ float tile[(TH + 2) * (TW + 2)];   // 340 floats
    const int t = threadIdx.x;
    const int bw = W_DIM / TW;                    // 15
    const int bx = blockIdx.x % bw;
    const int by = blockIdx.x / bw;
    const int w0p = bx * TW, h0p = by * TH;
    const int tx = t % TW, ty = t / TW;

    float acc = 0.f;
    for (int c = 0; c < CIMG + C2; ++c) {
        const float* src = (c < CIMG) ? (enh + (size_t)c * HW)
                                      : (pt_img + (size_t)(c - CIMG) * HW);
        for (int l = t; l < (TH + 2) * (TW + 2); l += 256) {
            int r = l / (TW + 2), cc = l - r * (TW + 2);
            int gh = h0p - 1 + r, gw = w0p - 1 + cc;
            float v = 0.f;
            if (gh >= 0 && gh < H_DIM && gw >= 0 && gw < W_DIM) v = src[gh * W_DIM + gw];
            tile[l] = v;
        }
        __syncthreads();
        const float* wc = sbw + c * 9;
        float s = 0.f;
#pragma unroll
        for (int kh = 0; kh < 3; ++kh)
#pragma unroll
            for (int kw = 0; kw < 3; ++kw)
                s = fmaf(tile[(ty + kh) * (TW + 2) + tx + kw], wc[kh * 3 + kw], s);
        acc += s;
        __syncthreads();
    }
    float cv = acc + sbb[0];
    attn[(h0p + ty) * W_DIM + (w0p + tx)] = 1.f / (1.f + __expf(-cv));
}

// ---------------------------------------------------------------- out = enhanced * attn
__global__ __launch_bounds__(256) void final_gate_kernel(const float* __restrict__ enh,
                                                         const float* __restrict__ attn,
                                                         float* __restrict__ out) {
    int t = blockIdx.x * blockDim.x + threadIdx.x;
    size_t e = (size_t)t * 4;
    if (e >= (size_t)CIMG * HW) return;
    int p = (int)(e % HW);
    float4 a = *(const float4*)(enh + e);
    float4 s = *(const float4*)(attn + p);
    float4 o;
    o.x = a.x * s.x; o.y = a.y * s.y; o.z = a.z * s.z; o.w = a.w * s.w;
    *(float4*)(out + e) = o;
}

// ----------------------------------------------------------------
extern "C" void kernel_launch(void* const* d_in, const int* in_sizes, int n_in,
                              void* d_out, int out_size, void* d_ws, size_t ws_size,
                              hipStream_t stream) {
    const float* img_feat = (const float*)d_in[0];
    const float* seg_prob = (const float*)d_in[1];
    const float* vf0 = (const float*)d_in[2];
    const int*   g0  = (const int*)d_in[3];
    const float* vf1 = (const float*)d_in[4];
    const int*   g1  = (const int*)d_in[5];
    const float* vf2 = (const float*)d_in[6];
    const int*   g2  = (const int*)d_in[7];
    const float* w0  = (const float*)d_in[8];
    const float* b0  = (const float*)d_in[9];
    const float* w1  = (const float*)d_in[10];
    const float* b1  = (const float*)d_in[11];
    const float* sbw = (const float*)d_in[12];
    const float* sbb = (const float*)d_in[13];
    float* out = (float*)d_out;

    char* ws = (char*)d_ws;
    size_t off = 0;
    int* idx = (int*)(ws + off);
    off += ((size_t)3 * GCELLS * sizeof(int) + 255) & ~(size_t)255;
    float* pt_img = (float*)(ws + off);
    off += (size_t)CIMG * HW * sizeof(float);
    float* enh = (float*)(ws + off);
    off += (size_t)CIMG * HW * sizeof(float);
    float* attn = (float*)(ws + off);

    init_idx_kernel<<<(3 * GCELLS + 255) / 256, 256, 0, stream>>>(idx);
    scatter_kernel<<<(3 * NP + 255) / 256, 256, 0, stream>>>(g0, g1, g2, idx);
    enhanced_kernel<<<((CIMG * HW / 4) + 255) / 256, 256, 0, stream>>>(img_feat, seg_prob, enh);
    ptimg_wmma_kernel<<<HW / NPIX, 256, 0, stream>>>(vf0, vf1, vf2, idx, w0, b0, w1, b1, pt_img);
    conv_attn_kernel<<<(W_DIM / TW) * (H_DIM / TH), 256, 0, stream>>>(enh, pt_img, sbw, sbb, attn);
    final_gate_kernel<<<((CIMG * HW / 4) + 255) / 256, 256, 0, stream>>>(enh, attn, out);
}